// RMSELoss_12086037970980
// MI455X (gfx1250) — compile-verified
//
#include <hip/hip_runtime.h>
#include <math.h>

#define EARLY_W 0.7f
#define LATE_W  1.5f

typedef __attribute__((ext_vector_type(2))) float v2f;
typedef __attribute__((ext_vector_type(8))) float v8f;

constexpr int THREADS = 256;           // 8 wave32 waves per block
constexpr int BLOCKS  = 2048;
constexpr int WAVES   = THREADS / 32;

// Wave32 all-lane sum via the CDNA5 matrix pipe.
// A (16x4 f32): lane L holds s in VGPR0 -> A[L%16][K = (L<16 ? 0 : 2)], 0 in VGPR1.
// B (4x16 f32): all ones. D[m][n] = rowsum_m for every column n.
// D layout: lanes 0-15 hold M=0..7 (VGPR0..7), lanes 16-31 hold M=8..15.
// sum(d[0..7]) per lane + xor-16 shuffle = total over all 32 lanes.
// Requires EXEC all ones: callers must invoke with no divergence.
__device__ __forceinline__ float wave_reduce_wmma(float s) {
    v2f a; a[0] = s;    a[1] = 0.0f;
    v2f b; b[0] = 1.0f; b[1] = 1.0f;
    v8f c = {};
    v8f d = __builtin_amdgcn_wmma_f32_16x16x4_f32(
        /*neg_a=*/false, a, /*neg_b=*/false, b,
        /*c_mod=*/(short)0, c, /*reuse_a=*/false, /*reuse_b=*/false);
    float p = 0.0f;
#pragma unroll
    for (int i = 0; i < 8; ++i) p += d[i];
    p += __shfl_xor(p, 16, 32);   // combine M=0..7 half with M=8..15 half
    return p;                     // full wave sum, broadcast to every lane
}

__global__ __launch_bounds__(THREADS)
void rmse_partial_kernel(const float4* __restrict__ inp,
                         const float4* __restrict__ tgt,
                         double* __restrict__ partials,
                         long long nrows) {
    const long long tid    = (long long)blockIdx.x * blockDim.x + threadIdx.x;
    const long long stride = (long long)gridDim.x * blockDim.x;

    float acc = 0.0f;
    // N is a multiple of total threads and of 32 -> uniform trip count,
    // EXEC stays all-ones for the WMMA reduction below.
    for (long long r = tid; r < nrows; r += stride) {
        // Speculative stream-ahead prefetch (dropped silently if past the
        // mapping) -> global_prefetch_b8.
        __builtin_prefetch((const void*)(inp + r + 4 * stride), 0, 0);
        __builtin_prefetch((const void*)(tgt + r + 4 * stride), 0, 0);

        float4 a = inp[r];   // global_load_b128, 16B/lane fully coalesced
        float4 t = tgt[r];
        float dx = t.x - a.x;
        float dy = t.y - a.y;
        float dz = t.z - a.z;
        float dw = t.w - a.w;
        float w0 = (dx >= 0.0f) ? EARLY_W : LATE_W;  // column-0 weight
        acc = fmaf(dx * dx, w0, acc);
        acc = fmaf(dy, dy, acc);
        acc = fmaf(dz, dz, acc);
        acc = fmaf(dw, dw, acc);
    }

    // Wave-level reduction on the matrix pipe (EXEC all ones here).
    float wsum = wave_reduce_wmma(acc);

    __shared__ float lds[WAVES];
    const int lane = threadIdx.x & 31;
    const int wid  = threadIdx.x >> 5;
    if (lane == 0) lds[wid] = wsum;
    __syncthreads();
    if (threadIdx.x == 0) {
        float bsum = 0.0f;
#pragma unroll
        for (int i = 0; i < WAVES; ++i) bsum += lds[i];
        partials[blockIdx.x] = (double)bsum;  // one B64 store per block
    }
}

__global__ __launch_bounds__(256)
void rmse_finalize_kernel(const double* __restrict__ partials, int nparts,
                          float* __restrict__ out, double inv_count) {
    __shared__ double lds[256];
    double s = 0.0;
    for (int i = threadIdx.x; i < nparts; i += 256) s += partials[i];
    lds[threadIdx.x] = s;
    __syncthreads();
#pragma unroll
    for (int off = 128; off > 0; off >>= 1) {
        if (threadIdx.x < off) lds[threadIdx.x] += lds[threadIdx.x + off];
        __syncthreads();
    }
    if (threadIdx.x == 0) {
        out[0] = (float)sqrt(lds[0] * inv_count);
    }
}

extern "C" void kernel_launch(void* const* d_in, const int* in_sizes, int n_in,
                              void* d_out, int out_size, void* d_ws, size_t ws_size,
                              hipStream_t stream) {
    (void)n_in; (void)out_size;
    const float4* inp = (const float4*)d_in[0];   // inputs  [N, 4] f32
    const float4* tgt = (const float4*)d_in[1];   // targets [N, 4] f32
    float* out        = (float*)d_out;            // scalar f32
    double* partials  = (double*)d_ws;

    const long long total_elems = (long long)in_sizes[0];   // N*D
    const long long nrows       = total_elems / 4;

    int blocks = BLOCKS;
    // Guard: never exceed the workspace (one double per block).
    size_t max_blocks = ws_size / sizeof(double);
    if ((size_t)blocks > max_blocks && max_blocks > 0) blocks = (int)max_blocks;
    if (blocks < 1) blocks = 1;

    rmse_partial_kernel<<<blocks, THREADS, 0, stream>>>(inp, tgt, partials, nrows);
    rmse_finalize_kernel<<<1, 256, 0, stream>>>(partials, blocks, out,
                                                1.0 / (double)total_elems);
}